// VectorQuantizer_57999238365708
// MI455X (gfx1250) — compile-verified
//
#include <hip/hip_runtime.h>

typedef float v2f __attribute__((ext_vector_type(2)));
typedef float v8f __attribute__((ext_vector_type(8)));

#define NUM_E   1024
#define DIM     256
#define SPW     4096          // 16*16*16 spatial sites per batch
#define NTOT    32768         // 8 * SPW
#define QELEMS  8388608       // NTOT * DIM

// ---------------------------------------------------------------------------
// Kernel 1: ||e_k||^2 per codebook row
// ---------------------------------------------------------------------------
__global__ void vq_norms(const float* __restrict__ cb, float* __restrict__ e2) {
    __shared__ float red[256];
    const int k = blockIdx.x;
    const int t = threadIdx.x;
    const float v = cb[k * DIM + t];
    red[t] = v * v;
    __syncthreads();
    for (int off = 128; off > 0; off >>= 1) {
        if (t < off) red[t] += red[t + off];
        __syncthreads();
    }
    if (t == 0) e2[k] = red[0];
}

// ---------------------------------------------------------------------------
// Kernel 2: fused fp32-WMMA GEMM + running argmin.
// One wave = one 16-row strip of z_flat. LDS layout: L[row][d], row=0..15.
// Score tile layout (16x16 f32 C/D): VGPR i, lane l -> row (i + 8*(l>>4)),
// col (l & 15).
// ---------------------------------------------------------------------------
__global__ __launch_bounds__(64) void vq_argmin(const float* __restrict__ z,
                                                const float* __restrict__ cb,
                                                const float* __restrict__ e2,
                                                int* __restrict__ idx) {
    __shared__ float lds[2][16 * DIM];          // 32 KB / block (2 waves)
    const int lane  = threadIdx.x & 31;
    const int wave  = threadIdx.x >> 5;
    const int strip = blockIdx.x * 2 + wave;    // 2048 strips total
    const int n0    = strip * 16;
    const int b     = n0 >> 12;                 // batch
    const int s0    = n0 & 4095;                // spatial base (strip-aligned)
    const float* zb = z + (size_t)b * (DIM * SPW) + s0;
    float* L = lds[wave];

    const int half = lane >> 4;                 // 0/1: which K-pair / row half
    const int rc   = lane & 15;                 // A-row / B-column for this lane

    // Stage strip: L[r*DIM + d] = z[b][d][s0 + r]  (coalesced 16-float runs)
    for (int it = 0; it < 128; ++it) {
        const int d = it * 2 + half;
        L[rc * DIM + d] = zb[(size_t)d * SPW + rc];
    }
    __syncthreads();

    float bestD[8];
    int   bestI[8];
#pragma unroll
    for (int i = 0; i < 8; ++i) { bestD[i] = 3.4e38f; bestI[i] = 0; }

    for (int kt = 0; kt < 64; ++kt) {
        const int kbase = kt * 16;
        const float* cbrow = cb + (size_t)(kbase + rc) * DIM; // this lane's code col
        const float e2k = e2[kbase + rc];

        v8f c = {0.f, 0.f, 0.f, 0.f, 0.f, 0.f, 0.f, 0.f};
#pragma unroll 8
        for (int d = 0; d < DIM; d += 4) {
            const int dbase = d + 2 * half;
            v2f a, bf;
            a.x  = L[rc * DIM + dbase];          // A[M=rc][K=dbase]
            a.y  = L[rc * DIM + dbase + 1];      // A[M=rc][K=dbase+1]
            bf.x = cbrow[dbase];                 // B[K=dbase][N=rc]
            bf.y = cbrow[dbase + 1];             // B[K=dbase+1][N=rc]
            // D = A(16x4) * B(4x16) + C   -- v_wmma_f32_16x16x4_f32
            c = __builtin_amdgcn_wmma_f32_16x16x4_f32(
                    false, a, false, bf, (short)0, c, false, false);
        }
#pragma unroll
        for (int i = 0; i < 8; ++i) {
            const float dist = e2k - 2.0f * c[i];   // ||z||^2 term is row-const
            if (dist < bestD[i]) { bestD[i] = dist; bestI[i] = kbase + rc; }
        }
    }

    // Cross-lane argmin within each 16-lane half; smaller index wins ties
#pragma unroll
    for (int i = 0; i < 8; ++i) {
        float d0 = bestD[i];
        int   i0 = bestI[i];
        for (int m = 1; m < 16; m <<= 1) {
            const float od = __shfl_xor(d0, m, 32);
            const int   oi = __shfl_xor(i0, m, 32);
            if (od < d0 || (od == d0 && oi < i0)) { d0 = od; i0 = oi; }
        }
        if (rc == 0) idx[n0 + i + 8 * half] = i0;
    }
}

// ---------------------------------------------------------------------------
// Kernel 3: gather quantized vectors (straight-through value == codebook row),
// write idx_out, accumulate sum((q - z)^2)
// ---------------------------------------------------------------------------
__global__ void vq_gather(const float* __restrict__ z,
                          const float* __restrict__ cb,
                          const int* __restrict__ idx,
                          float* __restrict__ out,
                          float* __restrict__ acc) {
    __shared__ float red[256];
    const size_t flat = (size_t)blockIdx.x * 256 + threadIdx.x;
    const int s = (int)(flat & 4095);
    const int d = (int)((flat >> 12) & 255);
    const int b = (int)(flat >> 20);
    const int n = b * SPW + s;
    const int k = idx[n];
    const float q  = cb[(size_t)k * DIM + d];
    const float zv = z[flat];
    out[flat] = q;                               // quant_out (B,D,H,W,L)
    if (d == 0) out[(size_t)QELEMS + n] = (float)k;  // idx_out region
    const float diff = q - zv;
    red[threadIdx.x] = diff * diff;
    __syncthreads();
    for (int off = 128; off > 0; off >>= 1) {
        if (threadIdx.x < off) red[threadIdx.x] += red[threadIdx.x + off];
        __syncthreads();
    }
    if (threadIdx.x == 0) atomicAdd(acc, red[0]);
}

__global__ void vq_init(float* __restrict__ acc) { *acc = 0.0f; }

__global__ void vq_loss(const float* __restrict__ acc, float* __restrict__ out) {
    // q_latent + 0.25 * e_latent, both numerically mean((q - z)^2)
    out[(size_t)QELEMS + NTOT] = 1.25f * (*acc) / (float)QELEMS;
}

// ---------------------------------------------------------------------------
extern "C" void kernel_launch(void* const* d_in, const int* in_sizes, int n_in,
                              void* d_out, int out_size, void* d_ws, size_t ws_size,
                              hipStream_t stream) {
    const float* z  = (const float*)d_in[0];   // (8,256,16,16,16) fp32
    const float* cb = (const float*)d_in[1];   // (1024,256) fp32
    float* out  = (float*)d_out;               // quant | idx | loss
    float* ws_f = (float*)d_ws;
    float* e2   = ws_f;                        // 1024 floats
    int*   idx  = (int*)(ws_f + NUM_E);        // 32768 ints
    float* acc  = ws_f + NUM_E + NTOT;         // 1 float

    vq_init  <<<1,     1,   0, stream>>>(acc);
    vq_norms <<<NUM_E, 256, 0, stream>>>(cb, e2);
    vq_argmin<<<1024,  64,  0, stream>>>(z, cb, e2, idx);
    vq_gather<<<QELEMS / 256, 256, 0, stream>>>(z, cb, idx, out, acc);
    vq_loss  <<<1,     1,   0, stream>>>(acc, out);
}